// CriticGCN_12446815223936
// MI455X (gfx1250) — compile-verified
//
#include <hip/hip_runtime.h>

// Problem constants (from reference)
#define BATCH 16
#define NNODE 50000
#define NEDGE 1600000
#define FIN   2
#define FHID  8

typedef __attribute__((ext_vector_type(2))) float v2f;
typedef __attribute__((ext_vector_type(8))) float v8f;

// ---------------------------------------------------------------------------
// Kernel 1: init deg=1.0 (self-loop) and zero the 16x8 output accumulator.
// ---------------------------------------------------------------------------
__global__ void k_init(float* __restrict__ deg, float* __restrict__ out) {
  int i = blockIdx.x * blockDim.x + threadIdx.x;
  deg[i] = 1.0f;                       // exactly B*N threads launched
  if (i < BATCH * FHID) out[i] = 0.0f;
}

// ---------------------------------------------------------------------------
// Kernel 2: in-degree count over target (col) nodes. Streaming read of the
// second half of each batch's edge list; HW float atomics (no-return ->
// STOREcnt path, no CAS loop).
// ---------------------------------------------------------------------------
__global__ void k_degree(const int* __restrict__ ei, float* __restrict__ deg) {
  int i = blockIdx.x * blockDim.x + threadIdx.x;   // exactly B*E threads
  int b = i / NEDGE;
  int e = i - b * NEDGE;
  int col = ei[(size_t)b * 2 * NEDGE + NEDGE + e];
  unsafeAtomicAdd(&deg[b * NNODE + col], 1.0f);
}

// ---------------------------------------------------------------------------
// Kernel 3: per-node prep: dinv = rsqrt(deg) (stored back in place),
// xw = x @ conv_w, and the self-loop contribution agg = xw * dinv^2
// (plain store; later edge kernel atomically accumulates on top).
// ---------------------------------------------------------------------------
__global__ void k_node_prep(const float* __restrict__ x,
                            const float* __restrict__ conv_w,
                            float* __restrict__ deg_dinv,
                            float* __restrict__ xw,
                            float* __restrict__ agg) {
  int i = blockIdx.x * blockDim.x + threadIdx.x;   // exactly B*N threads
  float d = deg_dinv[i];                            // >= 1 always
  float dinv = rsqrtf(d);
  deg_dinv[i] = dinv;
  float x0 = x[2 * i], x1 = x[2 * i + 1];
  float w00 = conv_w[0], w01 = conv_w[1], w10 = conv_w[2], w11 = conv_w[3];
  float xw0 = x0 * w00 + x1 * w10;
  float xw1 = x0 * w01 + x1 * w11;
  xw[2 * i]     = xw0;
  xw[2 * i + 1] = xw1;
  float s = dinv * dinv;
  agg[2 * i]     = xw0 * s;
  agg[2 * i + 1] = xw1 * s;
}

// ---------------------------------------------------------------------------
// Kernel 4: edge scatter: agg[col] += xw[row] * dinv[row]*dinv[col].
// This is the bandwidth-dominant pass (row+col index stream + gathers +
// 2 HW f32 atomics/edge).
// ---------------------------------------------------------------------------
__global__ void k_edge(const int* __restrict__ ei,
                       const float* __restrict__ dinv,
                       const float* __restrict__ xw,
                       float* __restrict__ agg) {
  int i = blockIdx.x * blockDim.x + threadIdx.x;   // exactly B*E threads
  int b = i / NEDGE;
  int e = i - b * NEDGE;
  const int* p = ei + (size_t)b * 2 * NEDGE;
  int row = p[e];
  int col = p[e + NEDGE];
  float nr = dinv[b * NNODE + row] * dinv[b * NNODE + col];
  size_t src = (size_t)(b * NNODE + row) * 2;
  size_t dst = (size_t)(b * NNODE + col) * 2;
  unsafeAtomicAdd(&agg[dst],     xw[src]     * nr);
  unsafeAtomicAdd(&agg[dst + 1], xw[src + 1] * nr);
}

// ---------------------------------------------------------------------------
// Kernel 5: fused  h = relu(agg+b)+x ; leaky(h@W1+b1) ; leaky(h1@W2+b2) ;
// sum-pool. One wave handles a 16-node tile via V_WMMA_F32_16X16X4_F32:
//   lin1: D1 = A(16x4: nodes x [h0,h1,0,0]) * B(4x16: W1 zero-padded) + bias
//   lin2: two chained K=4 WMMAs (K=0..3, K=4..7) accumulating through C.
// D-layout -> A-layout transpose for lin2 goes through 512B of LDS per wave
// with an explicit s_wait_dscnt 0 (intra-wave DS ops are in-order otherwise).
// N = 50000 = 3125 * 16 exactly, so EXEC is all-1s at every WMMA.
// ---------------------------------------------------------------------------
__global__ __launch_bounds__(256) void k_mlp_pool(
    const float* __restrict__ x, const float* __restrict__ agg,
    const float* __restrict__ conv_b,
    const float* __restrict__ l1w, const float* __restrict__ l1b,
    const float* __restrict__ l2w, const float* __restrict__ l2b,
    float* __restrict__ out) {
  __shared__ float s_h1[8 * 16 * FHID];   // 8 waves * 16 nodes * 8 feats
  __shared__ float s_out[FHID];

  const int b    = blockIdx.y;
  const int lane = threadIdx.x & 31;
  const int wave = threadIdx.x >> 5;
  const int n16  = lane & 15;
  const int hi   = lane >> 4;             // lane group (0: lanes 0-15)

  // ---- B-matrix (4x16, K x N) for lin1: VGPR0 holds K=(hi?2:0), VGPR1 K=(hi?3:1)
  v2f b1;
  b1.x = (!hi && n16 < FHID) ? l1w[0 * FHID + n16] : 0.0f;  // K=0 row (K=2 is zero-pad)
  b1.y = (!hi && n16 < FHID) ? l1w[1 * FHID + n16] : 0.0f;  // K=1 row (K=3 is zero-pad)

  // ---- B-matrices for lin2 (8x8 in two K=4 slabs)
  v2f b2lo, b2hi;
  b2lo.x = (n16 < FHID) ? l2w[(2 * hi + 0) * FHID + n16] : 0.0f;  // K=0 / K=2
  b2lo.y = (n16 < FHID) ? l2w[(2 * hi + 1) * FHID + n16] : 0.0f;  // K=1 / K=3
  b2hi.x = (n16 < FHID) ? l2w[(4 + 2 * hi + 0) * FHID + n16] : 0.0f;  // K=4 / K=6
  b2hi.y = (n16 < FHID) ? l2w[(4 + 2 * hi + 1) * FHID + n16] : 0.0f;  // K=5 / K=7

  // ---- bias broadcast along M (C[M][N] = bias[N])
  const float bias1 = l1b[n16 & (FHID - 1)];
  const float bias2 = l2b[n16 & (FHID - 1)];
  v8f c1, c2;
#pragma unroll
  for (int v = 0; v < 8; ++v) { c1[v] = bias1; c2[v] = bias2; }

  const float cb0 = conv_b[0], cb1 = conv_b[1];
  const float* aggb = agg + (size_t)b * NNODE * 2;
  const float* xb   = x   + (size_t)b * NNODE * 2;
  float* sh = s_h1 + wave * (16 * FHID);

  const int tiles = NNODE / 16;                 // 3125, exact
  const int gw = blockIdx.x * 8 + wave;
  const int tw = gridDim.x * 8;

  v8f acc;
#pragma unroll
  for (int v = 0; v < 8; ++v) acc[v] = 0.0f;

  for (int t = gw; t < tiles; t += tw) {
    const int node = t * 16 + n16;
    // A for lin1: lanes 0-15 carry [h0,h1] (K=0,1); lanes 16-31 carry K=2,3 = 0
    v2f a1;
    if (!hi) {
      float h0 = aggb[2 * node]     + cb0; h0 = h0 > 0.0f ? h0 : 0.0f; h0 += xb[2 * node];
      float h1 = aggb[2 * node + 1] + cb1; h1 = h1 > 0.0f ? h1 : 0.0f; h1 += xb[2 * node + 1];
      a1.x = h0; a1.y = h1;
    } else {
      a1.x = 0.0f; a1.y = 0.0f;
    }

    v8f d1 = __builtin_amdgcn_wmma_f32_16x16x4_f32(
        false, a1, false, b1, (short)0, c1, false, false);
#pragma unroll
    for (int v = 0; v < 8; ++v) { float u = d1[v]; d1[v] = u > 0.0f ? u : 0.01f * u; }

    // Stage h1 (D-layout: lane holds [M = v+8*hi][N = n16]) into LDS row-major
    if (n16 < FHID) {
#pragma unroll
      for (int v = 0; v < 8; ++v) sh[(v + 8 * hi) * FHID + n16] = d1[v];
    }
    asm volatile("s_wait_dscnt 0x0" ::: "memory");

    // Reload in A-layout for lin2: lane -> row m=n16, K pair (2*hi, 2*hi+1)
    const int m = n16;
    v2f a2lo, a2hi;
    a2lo.x = sh[m * FHID + 2 * hi + 0];
    a2lo.y = sh[m * FHID + 2 * hi + 1];
    a2hi.x = sh[m * FHID + 4 + 2 * hi + 0];
    a2hi.y = sh[m * FHID + 4 + 2 * hi + 1];

    v8f d2 = __builtin_amdgcn_wmma_f32_16x16x4_f32(
        false, a2lo, false, b2lo, (short)0, c2, false, false);
    d2 = __builtin_amdgcn_wmma_f32_16x16x4_f32(
        false, a2hi, false, b2hi, (short)0, d2, false, false);

#pragma unroll
    for (int v = 0; v < 8; ++v) {
      float u = d2[v];
      u = u > 0.0f ? u : 0.01f * u;   // leaky_relu
      acc[v] += u;                     // pool over the 8 nodes this lane covers
    }
  }

  // Block reduction: lane partial covers feature n16 (valid for n16 < 8)
  if (threadIdx.x < FHID) s_out[threadIdx.x] = 0.0f;
  __syncthreads();
  if (n16 < FHID) {
    float p = acc[0] + acc[1] + acc[2] + acc[3] + acc[4] + acc[5] + acc[6] + acc[7];
    atomicAdd(&s_out[n16], p);        // ds_add_f32
  }
  __syncthreads();
  if (threadIdx.x < FHID)
    unsafeAtomicAdd(&out[b * FHID + threadIdx.x], s_out[threadIdx.x]);
}

// ---------------------------------------------------------------------------
extern "C" void kernel_launch(void* const* d_in, const int* in_sizes, int n_in,
                              void* d_out, int out_size, void* d_ws, size_t ws_size,
                              hipStream_t stream) {
  const float* x      = (const float*)d_in[0];
  const int*   ei     = (const int*)d_in[1];   // harness: integer -> int32
  const float* conv_w = (const float*)d_in[2];
  const float* conv_b = (const float*)d_in[3];
  const float* l1w    = (const float*)d_in[4];
  const float* l1b    = (const float*)d_in[5];
  const float* l2w    = (const float*)d_in[6];
  const float* l2b    = (const float*)d_in[7];
  float* out = (float*)d_out;

  // Workspace layout (floats): deg/dinv [B*N] | xw [2*B*N] | agg [2*B*N]
  float* ws   = (float*)d_ws;
  float* deg  = ws;
  float* xw   = ws + (size_t)BATCH * NNODE;
  float* agg  = ws + (size_t)3 * BATCH * NNODE;

  const int node_blocks = (BATCH * NNODE) / 256;   // 3125, exact
  const int edge_blocks = (BATCH * NEDGE) / 256;   // 100000, exact

  k_init<<<node_blocks, 256, 0, stream>>>(deg, out);
  k_degree<<<edge_blocks, 256, 0, stream>>>(ei, deg);
  k_node_prep<<<node_blocks, 256, 0, stream>>>(x, conv_w, deg, xw, agg);
  k_edge<<<edge_blocks, 256, 0, stream>>>(ei, deg, xw, agg);
  k_mlp_pool<<<dim3(25, BATCH), 256, 0, stream>>>(
      x, agg, conv_b, l1w, l1b, l2w, l2b, out);
}